// EuclideanCodebook_67302137528709
// MI455X (gfx1250) — compile-verified
//
#include <hip/hip_runtime.h>
#include <hip/hip_bf16.h>

// Problem constants (match reference)
#define BB 16
#define TT 2048
#define DD 128
#define KK 8192

#define NROWS (BB * TT)          // 32768 query rows
#define NTILES (KK / 16)         // 512 code tiles of 16
#define LDS_STRIDE 132           // 128 + 4 pad -> conflict-free ds_load_b64

typedef float v2f __attribute__((ext_vector_type(2)));
typedef float v8f __attribute__((ext_vector_type(8)));

// -------------------------------------------------------------------------
// Pass 1: ||e_k||^2 for all K codes (one wave per code), + commit_loss = 0
// -------------------------------------------------------------------------
__global__ __launch_bounds__(256) void vq_norms(const float* __restrict__ embed,
                                                float* __restrict__ enorm,
                                                float* __restrict__ commit) {
    int lane = threadIdx.x & 31;
    int code = (blockIdx.x * blockDim.x + threadIdx.x) >> 5;
    if (code < KK) {
        const float4 v = *(const float4*)(embed + (size_t)code * DD + lane * 4);
        float s = v.x * v.x + v.y * v.y + v.z * v.z + v.w * v.w;
#pragma unroll
        for (int m = 16; m >= 1; m >>= 1) s += __shfl_xor(s, m, 32);
        if (lane == 0) enorm[code] = s;
    }
    if (blockIdx.x == 0 && threadIdx.x == 0) commit[0] = 0.0f;
}

// -------------------------------------------------------------------------
// Stage one 16x128 fp32 codebook tile into padded LDS (256 threads, 32 B each)
// -------------------------------------------------------------------------
__device__ __forceinline__ void stage_tile(float* __restrict__ dst,
                                           const float* __restrict__ embed,
                                           int ctile, int tid) {
    const int code = tid >> 4;           // 0..15
    const int dOff = (tid & 15) * 8;     // 0..120
    const float4* src =
        (const float4*)(embed + ((size_t)(ctile * 16 + code) * DD + dOff));
    float4 a = src[0];
    float4 b = src[1];
    float4* d = (float4*)(dst + code * LDS_STRIDE + dOff);
    d[0] = a;
    d[1] = b;
}

// -------------------------------------------------------------------------
// Pass 2: per-row argmax over K via fp32 WMMA, then fused gather + index out
//   block = 256 threads (8 waves), block owns 128 rows, wave owns 16 rows
// -------------------------------------------------------------------------
__global__ __launch_bounds__(256) void vq_argmax(const float* __restrict__ x,
                                                 const float* __restrict__ embed,
                                                 const float* __restrict__ enorm,
                                                 float* __restrict__ out_q,
                                                 float* __restrict__ out_ind) {
    __shared__ __align__(16) float eTile[2][16 * LDS_STRIDE];

    const int tid   = threadIdx.x;
    const int wave  = tid >> 5;
    const int lane  = tid & 31;
    const int lhalf = lane >> 4;   // 0 or 1 -> K-pair select / row-group select
    const int lmod  = lane & 15;   // A: row within tile; B/C/D: column (code)

    const int waveRow = blockIdx.x * 128 + wave * 16;   // first of 16 rows
    const int myRow   = waveRow + lmod;

    // ---- Preload A fragments for the whole D=128 reduction (32 x v2f) ----
    // 16x4 fp32 A layout: lane L holds row L%16, K-pair at 2*(L/16)
    v2f areg[32];
    const float* xrow = x + (size_t)myRow * DD;
#pragma unroll
    for (int t = 0; t < 32; ++t) {
        const int d = 4 * t + 2 * lhalf;
        areg[t].x = xrow[d];
        areg[t].y = xrow[d + 1];
    }

    float best[8];
    int   bidx[8];
#pragma unroll
    for (int j = 0; j < 8; ++j) { best[j] = -3.402823466e38f; bidx[j] = 0; }

    stage_tile(&eTile[0][0], embed, 0, tid);

    for (int ct = 0; ct < NTILES; ++ct) {
        __syncthreads();                     // staged tile `ct` visible
        const int buf = ct & 1;
        if (ct + 1 < NTILES) stage_tile(&eTile[buf ^ 1][0], embed, ct + 1, tid);

        // ---- S = X_tile * E_tile^T  (16x16 fp32 via 32 chained WMMA) ----
        const float* ep = &eTile[buf][lmod * LDS_STRIDE + 2 * lhalf];
        v8f acc = {0.f, 0.f, 0.f, 0.f, 0.f, 0.f, 0.f, 0.f};
#pragma unroll
        for (int t = 0; t < 32; ++t) {
            v2f b = *(const v2f*)(ep + 4 * t);   // ds_load_b64, bank-conflict-free
            acc = __builtin_amdgcn_wmma_f32_16x16x4_f32(
                false, areg[t], false, b, (short)0, acc, false, false);
        }

        // ---- score = 2*S - ||e||^2 ; running argmax (first-index ties) ----
        const int   code = (ct << 4) + lmod;
        const float nrm  = enorm[code];
#pragma unroll
        for (int j = 0; j < 8; ++j) {
            const float sc = 2.0f * acc[j] - nrm;
            if (sc > best[j]) { best[j] = sc; bidx[j] = code; }
        }
    }

    // ---- reduce (score, idx) across the 16 lanes of each half-wave ----
#pragma unroll
    for (int j = 0; j < 8; ++j) {
#pragma unroll
        for (int m = 1; m <= 8; m <<= 1) {
            const float os = __shfl_xor(best[j], m, 32);
            const int   oi = __shfl_xor(bidx[j], m, 32);
            if (os > best[j] || (os == best[j] && oi < bidx[j])) {
                best[j] = os;
                bidx[j] = oi;
            }
        }
    }

    // ---- fused gather: whole wave copies each winning 512B embed row ----
#pragma unroll 4
    for (int r = 0; r < 16; ++r) {
        // row r lives in slot (r&7) of lanes [(r>>3)*16 .. +15]
        const int w = __shfl(bidx[r & 7], (r >> 3) << 4, 32);
        const float4* src = (const float4*)(embed + (size_t)w * DD) + lane;
        float4* dst = (float4*)(out_q + (size_t)(waveRow + r) * DD) + lane;
        *dst = *src;
        if (lane == 0) out_ind[waveRow + r] = (float)w;
    }
}

// -------------------------------------------------------------------------
extern "C" void kernel_launch(void* const* d_in, const int* in_sizes, int n_in,
                              void* d_out, int out_size, void* d_ws, size_t ws_size,
                              hipStream_t stream) {
    const float* x     = (const float*)d_in[0];
    // d_in[1] = x_len (int32) — unused by the eval-path reference
    const float* embed = (const float*)d_in[2];

    float* out      = (float*)d_out;
    float* out_q    = out;                               // B*T*D floats
    float* commit   = out + (size_t)BB * TT * DD;        // 1 float (0.0)
    float* out_ind  = commit + 1;                        // B*T indices (as float)

    float* enorm = (float*)d_ws;                         // K floats = 32 KB

    vq_norms<<<dim3(KK / 8), dim3(256), 0, stream>>>(embed, enorm, commit);
    vq_argmax<<<dim3(NROWS / 128), dim3(256), 0, stream>>>(x, embed, enorm,
                                                           out_q, out_ind);
}